// DIMPA_50491635532103
// MI455X (gfx1250) — compile-verified
//
#include <hip/hip_runtime.h>

// DIMPA 2-hop signed/directed propagation for MI455X (gfx1250).
// Memory-bound SpMM-style scatter/gather: optimized for the CDNA5 memory
// system (wave32: one wave == one 128-float row via float4/B128 accesses,
// f32 global atomics absorbed by the 192MB L2, async global->LDS staging).
//
// Workspace (floats): deg_s[N] deg_t[N] b1s[N*D] b1t[N*D] b2s[N*D] b2t[N*D]
//   = (2N + 4*N*D)*4 bytes  (~103 MB for N=50000, D=128)

#define DIMPA_D    128
#define DIMPA_FILL 0.5f

#if defined(__gfx1250__) && __has_builtin(__builtin_amdgcn_global_load_async_to_lds_b128) && __has_builtin(__builtin_amdgcn_s_wait_asynccnt)
#define DIMPA_ASYNC 1
#else
#define DIMPA_ASYNC 0
#endif

#if DIMPA_ASYNC
typedef int v4i __attribute__((ext_vector_type(4)));
typedef __attribute__((address_space(1))) v4i GV4;   // global (prints as __device__)
typedef __attribute__((address_space(3))) v4i LV4;   // LDS
#endif

// ---------------------------------------------------------------- utilities
__global__ void k_zero(float* __restrict__ p, int n) {
    int i = blockIdx.x * blockDim.x + threadIdx.x;
    if (i < n) p[i] = 0.0f;
}

// deg_s[i] = sum of w over edges with src==i ; deg_t[i] over dst==i
__global__ void k_deg(const int* __restrict__ src, const int* __restrict__ dst,
                      const float* __restrict__ ew,
                      float* __restrict__ deg_s, float* __restrict__ deg_t, int E) {
    int e = blockIdx.x * blockDim.x + threadIdx.x;
    if (e < E) {
        float we = ew[e];
        atomicAdd(&deg_s[src[e]], we);
        atomicAdd(&deg_t[dst[e]], we);
    }
}

// in-place: deg -> 1/(deg + FILL)   (self-loop weight included; always > 0)
__global__ void k_inv(float* __restrict__ deg, int n) {
    int i = blockIdx.x * blockDim.x + threadIdx.x;
    if (i < n) deg[i] = 1.0f / (deg[i] + DIMPA_FILL);
}

// ------------------------------------------------- per-row streaming kernels
// wave-per-row: lane handles float4 at column lane*4  (32*4 == D == 128)

// out[i, 0:D]   = w_s[0]*x_s[i]   ; out[i, D:2D] = w_t[0]*x_t[i]
// b1s[i] = FILL*inv_s[i]*x_s[i]   ; b1t[i] = FILL*inv_t[i]*x_t[i]  (self loops)
__global__ void k_rows_init(const float* __restrict__ xs, const float* __restrict__ xt,
                            const float* __restrict__ inv_s, const float* __restrict__ inv_t,
                            const float* __restrict__ wsv, const float* __restrict__ wtv,
                            float* __restrict__ out,
                            float* __restrict__ b1s, float* __restrict__ b1t, int N) {
    int wv = threadIdx.x >> 5, lane = threadIdx.x & 31;
    int i = blockIdx.x * 8 + wv;
    if (i >= N) return;
    int c = lane * 4;
    float4 vs = *(const float4*)(xs + (size_t)i * DIMPA_D + c);
    float4 vt = *(const float4*)(xt + (size_t)i * DIMPA_D + c);
    float w0s = wsv[0], w0t = wtv[0];
    float4 os = make_float4(w0s * vs.x, w0s * vs.y, w0s * vs.z, w0s * vs.w);
    float4 ot = make_float4(w0t * vt.x, w0t * vt.y, w0t * vt.z, w0t * vt.w);
    *(float4*)(out + (size_t)i * 2 * DIMPA_D + c) = os;
    *(float4*)(out + (size_t)i * 2 * DIMPA_D + DIMPA_D + c) = ot;
    float as = DIMPA_FILL * inv_s[i], at = DIMPA_FILL * inv_t[i];
    *(float4*)(b1s + (size_t)i * DIMPA_D + c) =
        make_float4(as * vs.x, as * vs.y, as * vs.z, as * vs.w);
    *(float4*)(b1t + (size_t)i * DIMPA_D + c) =
        make_float4(at * vt.x, at * vt.y, at * vt.z, at * vt.w);
}

// out[i,0:D] += wsv[h]*bs[i]; out[i,D:2D] += wtv[h]*bt[i];
// if ns: ns[i] = FILL*inv_s[i]*bs[i], nt[i] = FILL*inv_t[i]*bt[i] (next hop self loop)
__global__ void k_accum_selfinit(const float* __restrict__ bs, const float* __restrict__ bt,
                                 const float* __restrict__ inv_s, const float* __restrict__ inv_t,
                                 const float* __restrict__ wsv, const float* __restrict__ wtv,
                                 int h, float* __restrict__ out,
                                 float* __restrict__ ns, float* __restrict__ nt, int N) {
    int wv = threadIdx.x >> 5, lane = threadIdx.x & 31;
    int i = blockIdx.x * 8 + wv;
    if (i >= N) return;
    int c = lane * 4;
    float4 vs = *(const float4*)(bs + (size_t)i * DIMPA_D + c);
    float4 vt = *(const float4*)(bt + (size_t)i * DIMPA_D + c);
    float whs = wsv[h], wht = wtv[h];
    float* po = out + (size_t)i * 2 * DIMPA_D + c;
    float4 os = *(float4*)po;
    os.x += whs * vs.x; os.y += whs * vs.y; os.z += whs * vs.z; os.w += whs * vs.w;
    *(float4*)po = os;
    float* pq = po + DIMPA_D;
    float4 ot = *(float4*)pq;
    ot.x += wht * vt.x; ot.y += wht * vt.y; ot.z += wht * vt.z; ot.w += wht * vt.w;
    *(float4*)pq = ot;
    if (ns != nullptr) {
        float as = DIMPA_FILL * inv_s[i], at = DIMPA_FILL * inv_t[i];
        *(float4*)(ns + (size_t)i * DIMPA_D + c) =
            make_float4(as * vs.x, as * vs.y, as * vs.z, as * vs.w);
        *(float4*)(nt + (size_t)i * DIMPA_D + c) =
            make_float4(at * vt.x, at * vt.y, at * vt.z, at * vt.w);
    }
}

// --------------------------------------------------------- edge scatter pass
// wave-per-edge, both directions per edge:
//   s: next_s[dst] += (w/deg_s[src]) * prev_s[src]
//   t: next_t[src] += (w/deg_t[dst]) * prev_t[dst]
// Gathered rows are staged global->LDS with the CDNA5 async path (ASYNCcnt),
// double-buffered across the two directions.
__global__ void k_scatter(const float* __restrict__ prev_s, const float* __restrict__ prev_t,
                          const float* __restrict__ inv_s, const float* __restrict__ inv_t,
                          const int* __restrict__ src, const int* __restrict__ dst,
                          const float* __restrict__ ew,
                          float* __restrict__ next_s, float* __restrict__ next_t, int E) {
#if DIMPA_ASYNC
    __shared__ __align__(16) float stage[8][2][DIMPA_D];   // 8 KB: [wave][dir][feat]
#endif
    int wv = threadIdx.x >> 5, lane = threadIdx.x & 31;
    int e = blockIdx.x * 8 + wv;
    if (e >= E) return;
    int s = src[e], d = dst[e];
    float we = ew[e];
    float cs = we * inv_s[s];
    float ct = we * inv_t[d];
    int c = lane * 4;
    const float* ps = prev_s + (size_t)s * DIMPA_D + c;
    const float* pt = prev_t + (size_t)d * DIMPA_D + c;
    float* nsp = next_s + (size_t)d * DIMPA_D + c;
    float* ntp = next_t + (size_t)s * DIMPA_D + c;
    __builtin_prefetch(nsp, 0, 3);   // global_prefetch_b8: warm scatter target in L2
    __builtin_prefetch(ntp, 0, 3);
#if DIMPA_ASYNC
    __builtin_amdgcn_global_load_async_to_lds_b128((GV4*)ps, (LV4*)&stage[wv][0][c], 0, 0);
    __builtin_amdgcn_global_load_async_to_lds_b128((GV4*)pt, (LV4*)&stage[wv][1][c], 0, 0);
    __builtin_amdgcn_s_wait_asynccnt(1);
    float4 v = *(const float4*)&stage[wv][0][c];
    atomicAdd(nsp + 0, cs * v.x);
    atomicAdd(nsp + 1, cs * v.y);
    atomicAdd(nsp + 2, cs * v.z);
    atomicAdd(nsp + 3, cs * v.w);
    __builtin_amdgcn_s_wait_asynccnt(0);
    float4 u = *(const float4*)&stage[wv][1][c];
#else
    float4 v = *(const float4*)ps;
    atomicAdd(nsp + 0, cs * v.x);
    atomicAdd(nsp + 1, cs * v.y);
    atomicAdd(nsp + 2, cs * v.z);
    atomicAdd(nsp + 3, cs * v.w);
    float4 u = *(const float4*)pt;
#endif
    atomicAdd(ntp + 0, ct * u.x);
    atomicAdd(ntp + 1, ct * u.y);
    atomicAdd(ntp + 2, ct * u.z);
    atomicAdd(ntp + 3, ct * u.w);
}

// ------------------------------------------------------------------- launch
extern "C" void kernel_launch(void* const* d_in, const int* in_sizes, int n_in,
                              void* d_out, int out_size, void* d_ws, size_t ws_size,
                              hipStream_t stream) {
    const float* x_s = (const float*)d_in[0];
    const float* x_t = (const float*)d_in[1];
    const int*  eidx = (const int*)d_in[2];
    const float* ew  = (const float*)d_in[3];
    const float* w_s = (const float*)d_in[4];
    const float* w_t = (const float*)d_in[5];

    const int N = in_sizes[0] / DIMPA_D;
    const int E = in_sizes[3];
    const int* src = eidx;          // edge_index row 0
    const int* dst = eidx + E;      // edge_index row 1

    float* ws    = (float*)d_ws;
    float* deg_s = ws;                                  // N  (inv_s after k_inv)
    float* deg_t = ws + N;                              // N  (inv_t after k_inv)
    float* b1s   = ws + 2 * (size_t)N;                  // N*D
    float* b1t   = b1s + (size_t)N * DIMPA_D;           // N*D
    float* b2s   = b1t + (size_t)N * DIMPA_D;           // N*D
    float* b2t   = b2s + (size_t)N * DIMPA_D;           // N*D
    float* out   = (float*)d_out;

    const dim3 blk(256);
    const int gz = (2 * N + 255) / 256;   // thread-per-element over 2N
    const int ge = (E + 255) / 256;       // thread-per-edge
    const int gw = (N + 7) / 8;           // wave-per-row (8 waves/block)
    const int gs = (E + 7) / 8;           // wave-per-edge

    // degrees -> inverse degrees (with self-loop fill)
    k_zero<<<gz, blk, 0, stream>>>(deg_s, 2 * N);
    k_deg<<<ge, blk, 0, stream>>>(src, dst, ew, deg_s, deg_t, E);
    k_inv<<<gz, blk, 0, stream>>>(deg_s, 2 * N);

    // hop 0 contribution + hop-1 self loops
    k_rows_init<<<gw, blk, 0, stream>>>(x_s, x_t, deg_s, deg_t, w_s, w_t,
                                        out, b1s, b1t, N);
    // hop 1 edge messages
    k_scatter<<<gs, blk, 0, stream>>>(x_s, x_t, deg_s, deg_t, src, dst, ew,
                                      b1s, b1t, E);
    // accumulate hop 1, init hop-2 self loops
    k_accum_selfinit<<<gw, blk, 0, stream>>>(b1s, b1t, deg_s, deg_t, w_s, w_t, 1,
                                             out, b2s, b2t, N);
    // hop 2 edge messages
    k_scatter<<<gs, blk, 0, stream>>>(b1s, b1t, deg_s, deg_t, src, dst, ew,
                                      b2s, b2t, E);
    // accumulate hop 2 (no next hop)
    k_accum_selfinit<<<gw, blk, 0, stream>>>(b2s, b2t, deg_s, deg_t, w_s, w_t, 2,
                                             out, nullptr, nullptr, N);
}